// TopKGate_90366111908241
// MI455X (gfx1250) — compile-verified
//
#include <hip/hip_runtime.h>
#include <math.h>

#define NTOK   8192
#define DMODEL 4096
#define NEXP   64
#define LBPAD  68    // padded LDS row stride (floats) for the weight tile
#define KCH    64    // K-chunk staged per iteration
#define GBLK   128   // gemm block size (4 waves x 16 rows = 64 rows/block)

typedef __attribute__((ext_vector_type(2))) float v2f;
typedef __attribute__((ext_vector_type(8))) float v8f;
typedef int v4i_vs __attribute__((vector_size(16)));   // matches builtin param type
typedef __attribute__((address_space(1))) v4i_vs* glb_v4i_ptr;
typedef __attribute__((address_space(3))) v4i_vs* lds_v4i_ptr;

#if __has_builtin(__builtin_amdgcn_global_load_async_to_lds_b128)
#define STAGE_ASYNC 1
#else
#define STAGE_ASYNC 0
#endif

static __device__ __forceinline__ void wait_async_zero() {
#if STAGE_ASYNC
#if __has_builtin(__builtin_amdgcn_s_wait_asynccnt)
  __builtin_amdgcn_s_wait_asynccnt(0);
#else
  asm volatile("s_wait_asynccnt 0x0" ::: "memory");
#endif
#endif
}

// ---------------------------------------------------------------------------
// Kernel 1: logits[n][e] = sum_d x[n][d] * wg[e][d]  via V_WMMA_F32_16X16X4_F32
// Block = 128 thr (4 waves), 128 blocks. Each wave: one 16-row M tile x four
// 16-col N tiles. wg chunk [64 x KCH] double-buffered in LDS; staging uses
// GLOBAL_LOAD_ASYNC_TO_LDS_B128 when the toolchain exposes it.
// ---------------------------------------------------------------------------
__global__ __launch_bounds__(GBLK) void moe_gemm_logits(
    const float* __restrict__ x, const float* __restrict__ wg,
    float* __restrict__ logits)
{
  __shared__ float lb[2][NEXP * LBPAD];
  const int tid  = threadIdx.x;
  const int wave = tid >> 5;
  const int lane = tid & 31;
  const int l16  = lane & 15;
  const int koff = (lane >> 4) << 1;            // 0 (lanes 0-15) / 2 (16-31)
  const int rowBase = blockIdx.x * 64 + wave * 16;
  const float* aptr = x + (size_t)(rowBase + l16) * DMODEL + koff;

  // Stage wg[0..63][kbase..kbase+KCH) into lb[buf]. 8 x b128 per thread.
  auto stage = [&](int kbase, int buf) {
#pragma unroll
    for (int p = 0; p < 8; ++p) {
      const int e = p * 8 + (tid >> 4);
      const int o = (tid & 15) * 4;
#if STAGE_ASYNC
      __builtin_amdgcn_global_load_async_to_lds_b128(
          (glb_v4i_ptr)(wg + (size_t)e * DMODEL + kbase + o),
          (lds_v4i_ptr)(&lb[buf][e * LBPAD + o]),
          0, 0);
#else
      *(float4*)(&lb[buf][e * LBPAD + o]) =
          *(const float4*)(wg + (size_t)e * DMODEL + kbase + o);
#endif
    }
  };

  v8f acc[4];
  const v8f vzero = {0.f, 0.f, 0.f, 0.f, 0.f, 0.f, 0.f, 0.f};
#pragma unroll
  for (int t = 0; t < 4; ++t) acc[t] = vzero;

  stage(0, 0);
  int buf = 0;
  for (int k = 0; k < DMODEL; k += KCH) {
    wait_async_zero();        // this wave's staging landed in LDS
    __syncthreads();          // everyone's staging landed
    if (k + KCH < DMODEL) stage(k + KCH, buf ^ 1);  // prefetch next chunk
    const float* lbuf = &lb[buf][0];

#pragma unroll
    for (int km = 0; km < KCH; km += 16) {
      // A 16x4 layout: lanes 0-15 hold (K,K+1), lanes 16-31 hold (K+2,K+3)
      v2f a[4];
#pragma unroll
      for (int u = 0; u < 4; ++u)
        a[u] = *(const v2f*)(aptr + k + km + 4 * u);
      // B 4x16 layout mirrors A with N across lanes; batch all 16 fragments
      v2f b[4][4];
#pragma unroll
      for (int t = 0; t < 4; ++t)
#pragma unroll
        for (int u = 0; u < 4; ++u)
          b[t][u] = *(const v2f*)(&lbuf[(t * 16 + l16) * LBPAD + km + 4 * u + koff]);
      // 16 WMMAs, accumulator-interleaved (no back-to-back C dependency)
#pragma unroll
      for (int u = 0; u < 4; ++u)
#pragma unroll
        for (int t = 0; t < 4; ++t)
          acc[t] = __builtin_amdgcn_wmma_f32_16x16x4_f32(
              false, a[u], false, b[t][u], (short)0, acc[t], false, false);
    }
    buf ^= 1;
  }

  // C 16x16 f32 layout: VGPR r, lanes 0-15 -> M=r, lanes 16-31 -> M=8+r
  const int rowAdd = (lane >> 4) << 3;
#pragma unroll
  for (int t = 0; t < 4; ++t)
#pragma unroll
    for (int r = 0; r < 8; ++r)
      logits[(size_t)(rowBase + r + rowAdd) * NEXP + t * 16 + l16] = acc[t][r];
}

// ---------------------------------------------------------------------------
// Kernel 2: per-token softmax + top-2 (lowest-index tie-break). 1 wave/token.
// ---------------------------------------------------------------------------
__global__ __launch_bounds__(256) void moe_gate_topk(
    const float* __restrict__ logits, float* __restrict__ gates_full,
    float* __restrict__ out)
{
  const int tok  = (blockIdx.x * blockDim.x + threadIdx.x) >> 5;
  const int lane = threadIdx.x & 31;
  const float l0 = logits[(size_t)tok * NEXP + lane];
  const float l1 = logits[(size_t)tok * NEXP + lane + 32];

  float m = fmaxf(l0, l1);
#pragma unroll
  for (int off = 16; off > 0; off >>= 1) m = fmaxf(m, __shfl_xor(m, off, 32));
  const float e0 = __expf(l0 - m);
  const float e1 = __expf(l1 - m);
  float s = e0 + e1;
#pragma unroll
  for (int off = 16; off > 0; off >>= 1) s += __shfl_xor(s, off, 32);
  const float inv = 1.0f / s;
  gates_full[(size_t)tok * NEXP + lane]      = e0 * inv;
  gates_full[(size_t)tok * NEXP + lane + 32] = e1 * inv;

  // top-1: (value, index) max with lowest-index tie-break
  float v1 = l0; int i1 = lane;
  if (l1 > v1) { v1 = l1; i1 = lane + 32; }
#pragma unroll
  for (int off = 16; off > 0; off >>= 1) {
    const float ov = __shfl_xor(v1, off, 32);
    const int   oi = __shfl_xor(i1, off, 32);
    if (ov > v1 || (ov == v1 && oi < i1)) { v1 = ov; i1 = oi; }
  }
  // top-2: same reduction excluding index i1
  float v2 = -3.4028235e38f; int i2 = 1 << 30;
  if (lane != i1) { v2 = l0; i2 = lane; }
  if ((lane + 32) != i1 && (l1 > v2 || (l1 == v2 && (lane + 32) < i2))) {
    v2 = l1; i2 = lane + 32;
  }
#pragma unroll
  for (int off = 16; off > 0; off >>= 1) {
    const float ov = __shfl_xor(v2, off, 32);
    const int   oi = __shfl_xor(i2, off, 32);
    if (ov > v2 || (ov == v2 && oi < i2)) { v2 = ov; i2 = oi; }
  }

  float gk0 = __expf(v1 - m) * inv;
  float gk1 = __expf(v2 - m) * inv;
  float denom = fmaxf(gk0 + gk1, 1.1920929e-07f);  // finfo(f32).eps
  gk0 /= denom; gk1 /= denom;

  if (lane == 0) {
    out[1 + tok]        = gk0;
    out[1 + NTOK + tok] = gk1;
    int* oi32 = (int*)out;
    oi32[1 + 2 * NTOK + tok] = i1;
    oi32[1 + 3 * NTOK + tok] = i2;
  }
}

// ---------------------------------------------------------------------------
// Kernel 3: l_loss = sum_e me[e]*ce[e] * E/N^2. Single 1024-thread block.
// ---------------------------------------------------------------------------
__global__ __launch_bounds__(1024) void moe_loss(
    const float* __restrict__ gates_full, float* __restrict__ out)
{
  __shared__ float sme[1024];
  __shared__ float sce[1024];
  const int e  = threadIdx.x & 63;
  const int sl = threadIdx.x >> 6;     // 16 row-slices per expert
  const int* idx0 = (const int*)out + 1 + 2 * NTOK;
  float me = 0.f, ce = 0.f;
  for (int n = sl; n < NTOK; n += 16) {
    me += gates_full[(size_t)n * NEXP + e];
    ce += (idx0[n] == e) ? 1.f : 0.f;
  }
  sme[threadIdx.x] = me;
  sce[threadIdx.x] = ce;
  __syncthreads();
  if (sl == 0) {
    float tm = 0.f, tc = 0.f;
    for (int s2 = 0; s2 < 16; ++s2) { tm += sme[s2 * 64 + e]; tc += sce[s2 * 64 + e]; }
    sme[e] = tm * tc;
  }
  __syncthreads();
  if (threadIdx.x == 0) {
    float a = 0.f;
    for (int i = 0; i < 64; ++i) a += sme[i];
    out[0] = a * ((float)NEXP / ((float)NTOK * (float)NTOK));
  }
}

// ---------------------------------------------------------------------------
// Kernel 4: per-block token-order ranks + per-block expert histograms over the
// flattened [K*N] assignment array (k-major, matching reference cumsum order).
// ---------------------------------------------------------------------------
__global__ __launch_bounds__(256) void moe_loc_rank(
    const float* __restrict__ outF, int* __restrict__ localrank,
    int* __restrict__ blockhist)
{
  __shared__ int whist[8 * NEXP];
  const int* idx = (const int*)outF + 1 + 2 * NTOK;
  const int g    = blockIdx.x * 256 + threadIdx.x;
  const int e    = idx[g];
  const int lane = threadIdx.x & 31;
  const int wave = threadIdx.x >> 5;
  whist[threadIdx.x]       = 0;
  whist[256 + threadIdx.x] = 0;
  __syncthreads();
  int rank = 0, cnt = 0;
  for (int j = 0; j < 32; ++j) {
    const int ej = __shfl(e, j, 32);
    cnt  += (ej == e) ? 1 : 0;
    rank += (ej == e && j < lane) ? 1 : 0;
  }
  if (rank == 0) whist[wave * NEXP + e] = cnt;  // first lane of each expert
  __syncthreads();
  int br = rank;
  for (int w = 0; w < wave; ++w) br += whist[w * NEXP + e];
  localrank[g] = br;
  if (threadIdx.x < NEXP) {
    int h = 0;
    for (int w = 0; w < 8; ++w) h += whist[w * NEXP + threadIdx.x];
    blockhist[blockIdx.x * NEXP + threadIdx.x] = h;
  }
}

// ---------------------------------------------------------------------------
// Kernel 5: exclusive block scan per expert + final location write.
// ---------------------------------------------------------------------------
__global__ __launch_bounds__(1024) void moe_loc_write(
    const int* __restrict__ localrank, const int* __restrict__ blockhist,
    float* __restrict__ out)
{
  __shared__ int offs[64 * NEXP];   // 64 blocks x 64 experts
  const int t = threadIdx.x;
  if (t < NEXP) {
    int run = 0;
    for (int b = 0; b < 64; ++b) {
      offs[b * NEXP + t] = run;
      run += blockhist[b * NEXP + t];
    }
  }
  __syncthreads();
  const int* idx = (const int*)out + 1 + 2 * NTOK;
  int* loc = (int*)out + 1 + 4 * NTOK;
  for (int g = t; g < 2 * NTOK; g += 1024) {
    const int e = idx[g];
    loc[g] = offs[(g >> 8) * NEXP + e] + localrank[g];
  }
}

extern "C" void kernel_launch(void* const* d_in, const int* in_sizes, int n_in,
                              void* d_out, int out_size, void* d_ws, size_t ws_size,
                              hipStream_t stream)
{
  const float* x  = (const float*)d_in[0];
  const float* wg = (const float*)d_in[1];
  (void)in_sizes; (void)n_in; (void)out_size; (void)ws_size;  // top_k fixed at 2

  float* out       = (float*)d_out;
  float* logits    = (float*)d_ws;                         // 8192*64 f32 (2 MB)
  float* gates     = logits + (size_t)NTOK * NEXP;         // 8192*64 f32 (2 MB)
  int*   localrank = (int*)(gates + (size_t)NTOK * NEXP);  // 16384 i32
  int*   blockhist = localrank + 2 * NTOK;                 // 64*64 i32

  moe_gemm_logits<<<dim3(NTOK / 64), dim3(GBLK), 0, stream>>>(x, wg, logits);
  moe_gate_topk<<<dim3((NTOK * 32) / 256), dim3(256), 0, stream>>>(logits, gates, out);
  moe_loss<<<dim3(1), dim3(1024), 0, stream>>>(gates, out);
  moe_loc_rank<<<dim3((2 * NTOK) / 256), dim3(256), 0, stream>>>(out, localrank, blockhist);
  moe_loc_write<<<dim3(1), dim3(1024), 0, stream>>>(localrank, blockhist, out);
}